// BinaryMasking_59880434041170
// MI455X (gfx1250) — compile-verified
//
#include <hip/hip_runtime.h>

// ---------------------------------------------------------------------------
// BinaryMasking top-K for MI455X (gfx1250).
// One workgroup per batch row; the whole row of 32768 sort keys lives in LDS
// (128KB of the WGP's 320KB). Radix-select (256-bin, 4 passes) finds the K-th
// largest key; mask = key > thr with exact stable tie handling. src mask kept
// as an LDS bitvector so the tgt pass reads the conditional from LDS.
// HBM traffic ~= 1x read of each U_base + 1x write of each mask (roofline).
// ---------------------------------------------------------------------------

#define BT    512           // threads per block (16 wave32)
#define TT    16
#define TLAT  32
#define TLON  64
#define NTOK  32768         // T*LAT*LON
#define NB    512           // batch
#define EPSU  1e-6f
#define LOGC  -27.631021115928547f   // logf(1e-12)

#if defined(__HIP_DEVICE_COMPILE__) && __has_builtin(__builtin_amdgcn_global_load_async_to_lds_b128)
#define USE_ASYNC_LDS 1
#else
#define USE_ASYNC_LDS 0
#endif

__device__ __forceinline__ float clampu(float u) {
    return fminf(fmaxf(u, EPSU), 1.0f - EPSU);
}

// monotone float -> uint mapping (larger float => larger key)
__device__ __forceinline__ unsigned f2ord(float f) {
    unsigned u = __float_as_uint(f);
    return (u & 0x80000000u) ? ~u : (u | 0x80000000u);
}

#if USE_ASYNC_LDS
// Types matching the builtin prototype: pointer-to-16B-int-vector in AS1/AS3.
typedef int v4i_vs __attribute__((vector_size(4 * sizeof(int))));
typedef __attribute__((address_space(1))) v4i_vs gv4i;
typedef __attribute__((address_space(3))) v4i_vs lv4i;

__device__ __forceinline__ void wait_async() {
#if __has_builtin(__builtin_amdgcn_s_wait_asynccnt)
    __builtin_amdgcn_s_wait_asynccnt(0);
#else
    asm volatile("s_wait_asynccnt 0" ::: "memory");
#endif
}
#endif

// Compute sortable keys for one row into LDS kbuf.
// Strided pattern: thread t handles elements t*4 + k*2048 (coalesced b128).
__device__ void gen_keys(const float* __restrict__ grow,
                         const float* __restrict__ ct,
                         const float* __restrict__ cla,
                         const float* __restrict__ clo,
                         unsigned* __restrict__ kbuf,
                         const unsigned* __restrict__ bits,
                         int useCond, unsigned tid)
{
#if USE_ASYNC_LDS
    // Async global -> LDS (CDNA5 async path, ASYNCcnt tracked).
    for (unsigned k = 0; k < 16u; ++k) {
        unsigned e = tid * 4u + k * (BT * 4u);
        __builtin_amdgcn_global_load_async_to_lds_b128(
            (gv4i*)(grow + e), (lv4i*)(kbuf + e), 0, 0);
    }
    wait_async();   // each thread only reads LDS it loaded itself
#endif
    for (unsigned k = 0; k < 16u; ++k) {
        unsigned e = tid * 4u + k * (BT * 4u);
        float x0, x1, x2, x3;
#if USE_ASYNC_LDS
        uint4 raw = *(const uint4*)(kbuf + e);
        x0 = __uint_as_float(raw.x); x1 = __uint_as_float(raw.y);
        x2 = __uint_as_float(raw.z); x3 = __uint_as_float(raw.w);
#else
        float4 f = *(const float4*)(grow + e);
        x0 = f.x; x1 = f.y; x2 = f.z; x3 = f.w;
#endif
        float add = ct[e >> 11] + cla[(e >> 6) & 31u];
        float c0 = 0.f, c1 = 0.f, c2 = 0.f, c3 = 0.f;
        if (useCond) {
            unsigned w = bits[e >> 5] >> (e & 31u);   // 4-aligned: same word
            c0 = (w & 1u) ? 0.0f : LOGC;
            c1 = (w & 2u) ? 0.0f : LOGC;
            c2 = (w & 4u) ? 0.0f : LOGC;
            c3 = (w & 8u) ? 0.0f : LOGC;
        }
        const float* cl = clo + (e & 63u);
        uint4 r;
        r.x = f2ord(logf(clampu(x0)) + add + cl[0] + c0);
        r.y = f2ord(logf(clampu(x1)) + add + cl[1] + c1);
        r.z = f2ord(logf(clampu(x2)) + add + cl[2] + c2);
        r.w = f2ord(logf(clampu(x3)) + add + cl[3] + c3);
        *(uint4*)(kbuf + e) = r;
    }
}

// 4-pass MSB->LSB radix select on LDS keys. Returns threshold key.
// ctrl[0]=prefix, ctrl[1]=remaining K among ==prefix class, ctrl[2]=eq count.
__device__ unsigned radix_select(const unsigned* __restrict__ kbuf,
                                 unsigned* __restrict__ hist,
                                 unsigned* __restrict__ ctrl,
                                 unsigned K, unsigned tid,
                                 unsigned* KpOut, unsigned* eqOut)
{
    if (tid == 0) { ctrl[0] = 0u; ctrl[1] = K; ctrl[2] = 0u; }
    __syncthreads();
    for (int shift = 24; shift >= 0; shift -= 8) {
        for (unsigned j = tid; j < 256u; j += BT) hist[j] = 0u;
        __syncthreads();
        const unsigned prefix = ctrl[0];
        const unsigned hiMask = (shift == 24) ? 0u : (0xFFFFFFFFu << (shift + 8));
        const unsigned sh = (unsigned)shift;
        for (unsigned e = tid * 4u; e < NTOK; e += BT * 4u) {
            uint4 k4 = *(const uint4*)(kbuf + e);
            if ((k4.x & hiMask) == prefix) atomicAdd(&hist[(k4.x >> sh) & 255u], 1u);
            if ((k4.y & hiMask) == prefix) atomicAdd(&hist[(k4.y >> sh) & 255u], 1u);
            if ((k4.z & hiMask) == prefix) atomicAdd(&hist[(k4.z >> sh) & 255u], 1u);
            if ((k4.w & hiMask) == prefix) atomicAdd(&hist[(k4.w >> sh) & 255u], 1u);
        }
        __syncthreads();
        if (tid == 0) {
            unsigned Kp = ctrl[1];
            unsigned cum = 0; int v = 0;
            for (int b = 255; b >= 0; --b) {
                unsigned c = hist[b];
                if (cum + c >= Kp) { v = b; if (shift == 0) ctrl[2] = c; break; }
                cum += c;
            }
            ctrl[0] = prefix | ((unsigned)v << sh);
            ctrl[1] = Kp - cum;
        }
        __syncthreads();
    }
    unsigned thr = ctrl[0];
    *KpOut = ctrl[1];
    *eqOut = ctrl[2];
    __syncthreads();
    return thr;
}

__device__ void write_mask(const unsigned* __restrict__ kbuf,
                           unsigned* __restrict__ bits,
                           float* __restrict__ orow,
                           unsigned thr, int takeEq, int storeBits, unsigned tid)
{
    for (unsigned e = tid * 4u; e < NTOK; e += BT * 4u) {
        uint4 k4 = *(const uint4*)(kbuf + e);
        bool b0 = (k4.x > thr) || (takeEq && k4.x == thr);
        bool b1 = (k4.y > thr) || (takeEq && k4.y == thr);
        bool b2 = (k4.z > thr) || (takeEq && k4.z == thr);
        bool b3 = (k4.w > thr) || (takeEq && k4.w == thr);
        float4 m;
        m.x = b0 ? 1.0f : 0.0f; m.y = b1 ? 1.0f : 0.0f;
        m.z = b2 ? 1.0f : 0.0f; m.w = b3 ? 1.0f : 0.0f;
        *(float4*)(orow + e) = m;
        if (storeBits) {
            unsigned bs = (b0 ? 1u : 0u) | (b1 ? 2u : 0u) | (b2 ? 4u : 0u) | (b3 ? 8u : 0u);
            if (bs) atomicOr(&bits[e >> 5], bs << (e & 31u));
        }
    }
}

// Rare path: fewer tied-at-threshold slots than tied elements. Stable rule:
// take the Kp lowest-index equals (matches double-argsort rank of the ref).
__device__ void fix_ties(const unsigned* __restrict__ kbuf,
                         unsigned* __restrict__ bits,
                         unsigned* __restrict__ scan,
                         float* __restrict__ orow,
                         unsigned thr, unsigned Kp, int storeBits, unsigned tid)
{
    const unsigned CH = NTOK / BT;        // 64 contiguous elems per thread
    const unsigned base = tid * CH;
    unsigned cnt = 0;
    for (unsigned j = 0; j < CH; ++j) cnt += (kbuf[base + j] == thr) ? 1u : 0u;
    scan[tid] = cnt;
    __syncthreads();
    if (tid == 0) {
        unsigned run = 0;
        for (unsigned i = 0; i < BT; ++i) { unsigned c = scan[i]; scan[i] = run; run += c; }
    }
    __syncthreads();
    unsigned pos = scan[tid];
    for (unsigned j = 0; j < CH; ++j) {
        unsigned i = base + j;
        if (kbuf[i] == thr) {
            if (pos < Kp) {
                orow[i] = 1.0f;
                if (storeBits) atomicOr(&bits[i >> 5], 1u << (i & 31u));
            }
            ++pos;
        }
    }
    __syncthreads();
}

__device__ void do_phase(const unsigned* __restrict__ kbuf,
                         unsigned* __restrict__ hist,
                         unsigned* __restrict__ ctrl,
                         unsigned* __restrict__ scan,
                         unsigned* __restrict__ bits,
                         float* __restrict__ orow,
                         int K, int storeBits, unsigned tid)
{
    unsigned thr; int takeEq; bool ties = false; unsigned Kp = 0, eq = 0;
    if (K <= 0)           { thr = 0xFFFFFFFFu; takeEq = 0; }          // all zero
    else if (K >= NTOK)   { thr = 0u;          takeEq = 1; }          // all one
    else {
        thr = radix_select(kbuf, hist, ctrl, (unsigned)K, tid, &Kp, &eq);
        takeEq = (Kp == eq);
        ties   = (Kp < eq);
    }
    write_mask(kbuf, bits, orow, thr, takeEq, storeBits, tid);
    __syncthreads();
    if (ties) fix_ties(kbuf, bits, scan, orow, thr, Kp, storeBits, tid);
}

extern "C" __global__ void __launch_bounds__(BT)
binary_topk_kernel(const float* __restrict__ Urs, const float* __restrict__ Urt,
                   const float* __restrict__ Ubs, const float* __restrict__ Uts,
                   const float* __restrict__ Ulas, const float* __restrict__ Ulos,
                   const float* __restrict__ Ubt, const float* __restrict__ Utt,
                   const float* __restrict__ Ulat, const float* __restrict__ Ulot,
                   float* __restrict__ out)
{
    extern __shared__ unsigned char smem_raw[];
    unsigned* kbuf = (unsigned*)smem_raw;        // 32768 u32 = 128KB (row of keys)
    unsigned* bits = kbuf + NTOK;                // 1024 u32   (src mask bitvector)
    float*    term = (float*)(bits + NTOK / 32); // 112 f32    (t/lat/lon log terms)
    unsigned* hist = (unsigned*)(term + 112);    // 256 u32
    unsigned* scan = hist + 256;                 // 512 u32
    unsigned* ctrl = scan + BT;                  // 8 u32

    const unsigned row = blockIdx.x;
    const unsigned tid = threadIdx.x;
    const size_t   rb  = (size_t)row * NTOK;

    const int Ksrc = (int)floorf(clampu(Urs[row]) * (float)NTOK);
    const int Ktgt = (int)floorf(clampu(Urt[row]) * (float)NTOK);

    bits[tid] = 0u; bits[tid + BT] = 0u;

    float* ct  = term;
    float* cla = term + TT;
    float* clo = term + TT + TLAT;
    if (tid < TT)                    ct[tid]               = 0.5f * logf(clampu(Uts[row * TT   + tid]));
    else if (tid < TT + TLAT)        cla[tid - TT]         = 0.5f * logf(clampu(Ulas[row * TLAT + (tid - TT)]));
    else if (tid < TT + TLAT + TLON) clo[tid - TT - TLAT]  = 0.5f * logf(clampu(Ulos[row * TLON + (tid - TT - TLAT)]));
    __syncthreads();

    // Pull the tgt base row toward L2 while the src phase runs (global_prefetch_b8).
    __builtin_prefetch(Ubt + rb + (size_t)tid * 64u, 0, 1);
    __builtin_prefetch(Ubt + rb + (size_t)tid * 64u + 32u, 0, 1);

    // ---- src phase ----
    gen_keys(Ubs + rb, ct, cla, clo, kbuf, bits, /*useCond=*/0, tid);
    __syncthreads();
    do_phase(kbuf, hist, ctrl, scan, bits, out + rb, Ksrc, /*storeBits=*/1, tid);
    __syncthreads();

    // ---- tgt phase ----
    if (tid < TT)                    ct[tid]               = 0.5f * logf(clampu(Utt[row * TT   + tid]));
    else if (tid < TT + TLAT)        cla[tid - TT]         = 0.5f * logf(clampu(Ulat[row * TLAT + (tid - TT)]));
    else if (tid < TT + TLAT + TLON) clo[tid - TT - TLAT]  = 0.5f * logf(clampu(Ulot[row * TLON + (tid - TT - TLAT)]));
    __syncthreads();

    gen_keys(Ubt + rb, ct, cla, clo, kbuf, bits, /*useCond=*/1, tid);
    __syncthreads();
    do_phase(kbuf, hist, ctrl, scan, bits, out + (size_t)NB * NTOK + rb, Ktgt, /*storeBits=*/0, tid);
}

extern "C" void kernel_launch(void* const* d_in, const int* in_sizes, int n_in,
                              void* d_out, int out_size, void* d_ws, size_t ws_size,
                              hipStream_t stream)
{
    (void)in_sizes; (void)n_in; (void)out_size; (void)d_ws; (void)ws_size;
    const float* Urs  = (const float*)d_in[0];
    const float* Urt  = (const float*)d_in[1];
    const float* Ubs  = (const float*)d_in[2];
    const float* Uts  = (const float*)d_in[3];
    const float* Ulas = (const float*)d_in[4];
    const float* Ulos = (const float*)d_in[5];
    const float* Ubt  = (const float*)d_in[6];
    const float* Utt  = (const float*)d_in[7];
    const float* Ulat = (const float*)d_in[8];
    const float* Ulot = (const float*)d_in[9];
    float* out = (float*)d_out;

    const size_t shmem = (size_t)(NTOK + NTOK / 32 + 256 + BT + 8) * sizeof(unsigned)
                       + 112 * sizeof(float);   // ~139KB of the WGP's 320KB LDS

    binary_topk_kernel<<<dim3(NB), dim3(BT), shmem, stream>>>(
        Urs, Urt, Ubs, Uts, Ulas, Ulos, Ubt, Utt, Ulat, Ulot, out);
}